// NCCLoss_55138790146331
// MI455X (gfx1250) — compile-verified
//
#include <hip/hip_runtime.h>

// NCC loss for (2,1,160,160,160) f32 volumes, window 9, eps 1e-5.
// All 15 separable box-filter passes run on the CDNA5 matrix pipe as banded
// GEMMs with V_WMMA_F32_16X16X4_F32:
//   X axis:  out = in  x Band   (band on the right, A = data rows)
//   Y/Z ax:  out = Band x in    (band on the left,  B = data rows, x-coalesced)
// 6 WMMAs per 16x16 output tile (band half-width 4 -> K in [m0-4, m0+19]).

typedef __attribute__((ext_vector_type(2))) float v2f;
typedef __attribute__((ext_vector_type(8))) float v8f;

#define NX     160
#define NLINE  51200          // 2 * 160 * 160 lines along X
#define NVOX   8192000        // 2 * 160^3
#define SY     160            // y stride (elements)
#define SZ     25600          // z stride (elements)
#define NTILEW 32000          // 320 slices * 10 * 10 tiles (Y or Z pass)
#define EPSF   1e-5f
#define WVOL   729.0f

// ---------------- box filter along X via WMMA: D = A(data) x B(band) -------
__global__ __launch_bounds__(256) void boxx_wmma(const float* __restrict__ in,
                                                 float* __restrict__ out) {
    int wid  = (blockIdx.x * blockDim.x + threadIdx.x) >> 5;   // one wave = 16 lines
    if (wid >= NLINE / 16) return;                             // wave-uniform
    int lane = threadIdx.x & 31;
    int hi   = lane >> 4;
    int nl   = lane & 15;
    int m0   = wid << 4;
    const float* base = in + m0 * NX;

    for (int j0 = 0; j0 < NX; j0 += 16) {          // 10 output tiles per line-group
        v8f acc = {};
        for (int kb = j0 - 4; kb <= j0 + 16; kb += 4) {
            if (kb < 0 || kb >= NX) continue;      // zero padding (uniform branch)
            int kk = kb + (hi << 1);
            // A 16x4: lane nl holds row m0+nl, K = kk, kk+1
            v2f a;
            a.x = base[nl * NX + kk];
            a.y = base[nl * NX + kk + 1];
            // B 4x16 band: B[k][n] = (|k - n| <= 4)
            int ng = j0 + nl;
            int d0 = kk - ng;     d0 = d0 < 0 ? -d0 : d0;
            int d1 = kk + 1 - ng; d1 = d1 < 0 ? -d1 : d1;
            v2f b;
            b.x = (d0 <= 4) ? 1.0f : 0.0f;
            b.y = (d1 <= 4) ? 1.0f : 0.0f;
            acc = __builtin_amdgcn_wmma_f32_16x16x4_f32(
                false, a, false, b, (short)0, acc, false, false);
        }
        int colg  = j0 + nl;
        int rbase = m0 + (hi << 3);
        #pragma unroll
        for (int r = 0; r < 8; ++r)
            out[(rbase + r) * NX + colg] = acc[r];
    }
}

// ------- box filter along Y/Z via WMMA: D = A(band) x B(data rows) ---------
// ZAXIS=false: filter y. slice (y,x) for fixed o=(b*160+z), base = o*25600.
// ZAXIS=true:  filter z. slice (z,x) for fixed o=(b*160+y),
//              base = (o/160)*4096000 + (o%160)*160.
template <bool ZAXIS>
__global__ __launch_bounds__(256) void box_wmma_lm(const float* __restrict__ in,
                                                   float* __restrict__ out) {
    int w = (blockIdx.x * blockDim.x + threadIdx.x) >> 5;
    if (w >= NTILEW) return;                       // wave-uniform (exact grid)
    int lane = threadIdx.x & 31;
    int hi   = lane >> 4;
    int nl   = lane & 15;

    int o  = w / 100;                              // slice index [0,320)
    int t  = w - o * 100;
    int m0 = (t / 10) << 4;                        // filter-axis tile origin
    int x0 = (t % 10) << 4;                        // x tile origin

    int fs, base;
    if (ZAXIS) { fs = SZ; base = (o / 160) * 4096000 + (o % 160) * SY; }
    else       { fs = SY; base = o * SZ; }

    const float* bp = in + base + x0 + nl;         // lane's x column
    v8f acc = {};
    for (int kb = m0 - 4; kb <= m0 + 16; kb += 4) {
        if (kb < 0 || kb >= 160) continue;         // zero padding (uniform)
        int k0 = kb + (hi << 1);
        // A 16x4 band: A[m][k] = (|m - k| <= 4), m = m0 + nl
        int m  = m0 + nl;
        int d0 = m - k0;       d0 = d0 < 0 ? -d0 : d0;
        int d1 = m - (k0 + 1); d1 = d1 < 0 ? -d1 : d1;
        v2f a;
        a.x = (d0 <= 4) ? 1.0f : 0.0f;
        a.y = (d1 <= 4) ? 1.0f : 0.0f;
        // B 4x16 data: row k striped across lanes (x-contiguous, coalesced)
        v2f b;
        b.x = bp[k0 * fs];
        b.y = bp[(k0 + 1) * fs];
        acc = __builtin_amdgcn_wmma_f32_16x16x4_f32(
            false, a, false, b, (short)0, acc, false, false);
    }
    float* op = out + base + x0 + nl;
    #pragma unroll
    for (int r = 0; r < 8; ++r)
        op[(m0 + r + (hi << 3)) * fs] = acc[r];
}

// ---------------- elementwise centering: td^2, pd^2, td*pd -----------------
__global__ __launch_bounds__(256) void ncc_center(const float* __restrict__ t,
                                                  const float* __restrict__ p,
                                                  const float* __restrict__ st,
                                                  const float* __restrict__ sp,
                                                  float* __restrict__ u1,
                                                  float* __restrict__ u2,
                                                  float* __restrict__ u3) {
    int i = blockIdx.x * 256 + threadIdx.x;
    if (i >= NVOX) return;
    const float inv = 1.0f / (WVOL + EPSF);
    float td = t[i] - st[i] * inv;
    float pd = p[i] - sp[i] * inv;
    u1[i] = td * td;
    u2[i] = pd * pd;
    u3[i] = td * pd;
}

// ---------------- cc + global sum (wave32 shuffle + LDS + f32 atomic) ------
__global__ __launch_bounds__(256) void ncc_reduce(const float* __restrict__ vt,
                                                  const float* __restrict__ vp,
                                                  const float* __restrict__ cr,
                                                  float* __restrict__ sum) {
    __shared__ float smem[8];
    float acc = 0.0f;
    for (int i = blockIdx.x * blockDim.x + threadIdx.x; i < NVOX;
         i += gridDim.x * blockDim.x) {
        float a = vt[i] + EPSF;
        float b = vp[i] + EPSF;
        acc += cr[i] / sqrtf(a * b);
    }
    for (int off = 16; off > 0; off >>= 1)
        acc += __shfl_down(acc, off, 32);
    if ((threadIdx.x & 31) == 0) smem[threadIdx.x >> 5] = acc;
    __syncthreads();
    if (threadIdx.x == 0) {
        float s = 0.0f;
        for (int w = 0; w < 8; ++w) s += smem[w];
        atomicAdd(sum, s);
    }
}

__global__ void ncc_finalize(const float* __restrict__ sum,
                             float* __restrict__ out) {
    out[0] = -sum[0] * (1.0f / (float)NVOX);
}

// ---------------------------------------------------------------------------
extern "C" void kernel_launch(void* const* d_in, const int* in_sizes, int n_in,
                              void* d_out, int out_size, void* d_ws, size_t ws_size,
                              hipStream_t stream) {
    const float* t = (const float*)d_in[0];   // y_true
    const float* p = (const float*)d_in[1];   // y_pred

    float* w = (float*)d_ws;
    const size_t F = (size_t)NVOX;
    float* w0 = w;
    float* w1 = w + F;
    float* w2 = w + 2 * F;
    float* w3 = w + 3 * F;
    float* w4 = w + 4 * F;
    float* sum = w + 5 * F;                   // 1 float accumulator

    const int gx = (NLINE / 16) * 32 / 256;   // 400 blocks (X pass)
    const int gt = NTILEW * 32 / 256;         // 4000 blocks (Y/Z pass)
    const int gv = (NVOX + 255) / 256;        // 32000 blocks (elementwise)

    // pass 1: 3D box sums of t and p (for the means)
    boxx_wmma        <<<gx, 256, 0, stream>>>(t,  w0);
    box_wmma_lm<false><<<gt, 256, 0, stream>>>(w0, w1);
    box_wmma_lm<true> <<<gt, 256, 0, stream>>>(w1, w2);   // S3_t -> w2
    boxx_wmma        <<<gx, 256, 0, stream>>>(p,  w0);
    box_wmma_lm<false><<<gt, 256, 0, stream>>>(w0, w1);
    box_wmma_lm<true> <<<gt, 256, 0, stream>>>(w1, w3);   // S3_p -> w3

    // centering: u1=td^2 (w0), u2=pd^2 (w1), u3=td*pd (w4)
    ncc_center<<<gv, 256, 0, stream>>>(t, p, w2, w3, w0, w1, w4);

    // pass 2: 3D box sums of the centered products
    boxx_wmma        <<<gx, 256, 0, stream>>>(w0, w2);
    box_wmma_lm<false><<<gt, 256, 0, stream>>>(w2, w0);
    box_wmma_lm<true> <<<gt, 256, 0, stream>>>(w0, w2);   // var_t -> w2
    boxx_wmma        <<<gx, 256, 0, stream>>>(w1, w3);
    box_wmma_lm<false><<<gt, 256, 0, stream>>>(w3, w1);
    box_wmma_lm<true> <<<gt, 256, 0, stream>>>(w1, w3);   // var_p -> w3
    boxx_wmma        <<<gx, 256, 0, stream>>>(w4, w0);
    box_wmma_lm<false><<<gt, 256, 0, stream>>>(w0, w4);
    box_wmma_lm<true> <<<gt, 256, 0, stream>>>(w4, w0);   // cross -> w0

    // reduction: -mean(cross / sqrt((var_t+eps)(var_p+eps)))
    hipMemsetAsync(sum, 0, sizeof(float), stream);
    ncc_reduce  <<<1024, 256, 0, stream>>>(w2, w3, w0, sum);
    ncc_finalize<<<1, 1, 0, stream>>>(sum, (float*)d_out);
}